// SpectralFreqTimeConv1D_70918499991694
// MI455X (gfx1250) — compile-verified
//
#include <hip/hip_runtime.h>
#include <hip/hip_bf16.h>

// ---------------------------------------------------------------------------
// SpectralFreqTimeConv1D on gfx1250 (MI455X).
// Truncated-DFT-as-GEMM formulation, fp32 WMMA (v_wmma_f32_16x16x4_f32).
//   B=16, N=16384, Cin=Cout=64, T=256, M=2049 (padded to MP=2112 = 33*64).
// out[b,n,o] = sum_{i<M} alpha_i * Re( tmod[b,i] * (sum_c XF[b,i,c] W[i,c,o])
//                                      * e^{+2*pi*i*n*i/N} ),  alpha_0=1 else 2
// with XF = (unnormalized) truncated rfft of x; 1/N folded into stage 3.
// Twiddles: per-lane complex rotation recurrence u <- u * e^{-i*dTheta}
// (dTheta constant per lane), resynced exactly each 64-chunk via
// theta = 2*pi*((row*n) & 16383)/16384.
// ---------------------------------------------------------------------------

typedef float v2f __attribute__((ext_vector_type(2)));
typedef float v8f __attribute__((ext_vector_type(8)));

#define BB 16
#define NN 16384
#define CC 64
#define TT 256
#define MM 2049
#define MP 2112                    // 33 * 64, padded mode count
#define NMASK 16383u
#define THETA_SCALE 3.8349519697141029e-4f   // 2*pi / 16384
#define LSTR 68                    // padded LDS row stride (floats), 68 % 64 == 4

static __device__ __forceinline__ v8f wmma_f32(v2f a, v2f b, v8f c) {
  // 8-arg form: (neg_a, A, neg_b, B, c_mod, C, reuse_a, reuse_b)
  return __builtin_amdgcn_wmma_f32_16x16x4_f32(false, a, false, b, (short)0, c,
                                               false, false);
}

// e^{-i * 2*pi*phase/N} for masked integer phase
static __device__ __forceinline__ void eneg(unsigned phase, float* re, float* im) {
  float s, c;
  __sincosf((float)(phase & NMASK) * THETA_SCALE, &s, &c);
  *re = c;
  *im = -s;
}

// ---------------------------------------------------------------------------
// Kernel 1: tmod[b,i] = (t_emb[b,:].dense_real_w[i,:], t_emb[b,:].dense_imag_w[i,:])
// One 64-thread block per mode i (padded modes written as zero).
// ---------------------------------------------------------------------------
__global__ __launch_bounds__(64) void tmod_kernel(
    const float* __restrict__ t_emb, const float* __restrict__ dr,
    const float* __restrict__ di, float* __restrict__ tmod) {
  const int i = blockIdx.x;
  const int tid = threadIdx.x;
  if (i >= MM) {
    if (tid < BB) {
      tmod[((size_t)tid * MP + i) * 2 + 0] = 0.f;
      tmod[((size_t)tid * MP + i) * 2 + 1] = 0.f;
    }
    return;
  }
  __shared__ float sdr[TT], sdi[TT];
#pragma unroll
  for (int k = 0; k < 4; ++k) {
    sdr[tid + 64 * k] = dr[(size_t)i * TT + tid + 64 * k];
    sdi[tid + 64 * k] = di[(size_t)i * TT + tid + 64 * k];
  }
  __syncthreads();
  if (tid < BB) {
    float ar = 0.f, ai = 0.f;
    const float* te = t_emb + (size_t)tid * TT;
    for (int t = 0; t < TT; ++t) {
      float e = te[t];
      ar += e * sdr[t];
      ai += e * sdi[t];
    }
    tmod[((size_t)tid * MP + i) * 2 + 0] = ar;
    tmod[((size_t)tid * MP + i) * 2 + 1] = ai;
  }
}

// ---------------------------------------------------------------------------
// Kernel 2: forward truncated DFT.
//   XFr[b,i,c] = sum_n x[b,n,c] * cos(theta),  XFi = sum_n x * (-sin(theta))
// Block = 128 threads (4 waves); wave owns a 16-mode tile x 64 channels.
// x chunk staged TRANSPOSED in LDS (xs[c*68 + n]) so each lane's B pair
// {x[n],x[n+1]} at fixed c is one aligned ds_load_b64 (conflict-free).
// A operand from rotation recurrence (no per-step transcendentals).
// A layout (16x16x4 f32): lane<16 -> M=lane, K=0,1; lane>=16 -> M=lane-16, K=2,3.
// ---------------------------------------------------------------------------
__global__ __launch_bounds__(128) void fwd_dft_kernel(
    const float* __restrict__ x, float* __restrict__ XFr,
    float* __restrict__ XFi) {
  __shared__ float xs[CC * LSTR];  // 17.4 KB, transposed+padded
  const int b = blockIdx.y;
  const int wave = threadIdx.x >> 5;
  const int lane = threadIdx.x & 31;
  const int row = lane & 15;
  const int half = lane >> 4;
  const int i_tile = blockIdx.x * 64 + wave * 16;
  const unsigned irow = (unsigned)(i_tile + row);

  // per-K4-step rotation constant: w = e^{-i*2*pi*(4*irow)/N}
  float wr, wi;
  eneg(irow * 4u, &wr, &wi);

  v8f accr[4], acci[4];
#pragma unroll
  for (int ct = 0; ct < 4; ++ct) {
    accr[ct] = (v8f){0.f, 0.f, 0.f, 0.f, 0.f, 0.f, 0.f, 0.f};
    acci[ct] = (v8f){0.f, 0.f, 0.f, 0.f, 0.f, 0.f, 0.f, 0.f};
  }

  for (int n0 = 0; n0 < NN; n0 += 64) {
    __syncthreads();  // protect previous chunk before overwrite
    const float4* src = (const float4*)x + ((size_t)b * NN + n0) * CC / 4;
    for (int t = threadIdx.x; t < 64 * CC / 4; t += 128) {
      float4 v = src[t];                // coalesced: x[b, n0+n, c4..c4+3]
      const int n = t >> 4;
      const int c = (t & 15) * 4;
      xs[(c + 0) * LSTR + n] = v.x;
      xs[(c + 1) * LSTR + n] = v.y;
      xs[(c + 2) * LSTR + n] = v.z;
      xs[(c + 3) * LSTR + n] = v.w;
    }
    __syncthreads();

    // resync rotation state exactly at chunk start
    float u0r, u0i, u1r, u1i;
    eneg(irow * (unsigned)(n0 + 2 * half), &u0r, &u0i);
    eneg(irow * (unsigned)(n0 + 2 * half + 1), &u1r, &u1i);

    for (int kk = 0; kk < 64; kk += 4) {
      const v2f ac = {u0r, u1r};   // cos(theta_k0), cos(theta_k1)
      const v2f as = {u0i, u1i};   // -sin(theta_k0), -sin(theta_k1)
      const int nl = kk + 2 * half;
#pragma unroll
      for (int ct = 0; ct < 4; ++ct) {
        const float2 p = *(const float2*)(xs + (ct * 16 + row) * LSTR + nl);
        const v2f bv = {p.x, p.y};
        accr[ct] = wmma_f32(ac, bv, accr[ct]);
        acci[ct] = wmma_f32(as, bv, acci[ct]);
      }
      // u *= w  (advance theta by 2*pi*4*irow/N)
      const float t0r = u0r * wr - u0i * wi, t0i = u0r * wi + u0i * wr;
      const float t1r = u1r * wr - u1i * wi, t1i = u1r * wi + u1i * wr;
      u0r = t0r; u0i = t0i; u1r = t1r; u1i = t1i;
    }
  }

  // D layout: VGPR r -> lane<16: M=r, N=lane ; lane>=16: M=r+8, N=lane-16
  const int ig = i_tile + half * 8;
#pragma unroll
  for (int ct = 0; ct < 4; ++ct)
#pragma unroll
    for (int r = 0; r < 8; ++r) {
      const size_t idx = ((size_t)b * MP + ig + r) * CC + ct * 16 + row;
      XFr[idx] = accr[ct][r];
      XFi[idx] = acci[ct][r];
    }
}

// ---------------------------------------------------------------------------
// Kernel 3: per-mode complex channel mix + time modulation + rfft/irfft scaling.
// One 64-thread block per (i, b); thread = output channel o. Runs in place on
// XF (fully rewritten by kernel 2 each call, so graph replay is deterministic).
// Padded modes i>=2049 are zero-filled so kernel 4 needs no masking.
// ---------------------------------------------------------------------------
__global__ __launch_bounds__(64) void modemix_kernel(
    const float* __restrict__ Wr, const float* __restrict__ Wi,
    const float* __restrict__ tmod, float* __restrict__ Xr,
    float* __restrict__ Xi) {
  const int i = blockIdx.x;
  const int b = blockIdx.y;
  const int o = threadIdx.x;
  const size_t rowbase = ((size_t)b * MP + i) * CC;
  if (i >= MM) {  // zero padded modes (no OOB weight reads)
    Xr[rowbase + o] = 0.f;
    Xi[rowbase + o] = 0.f;
    return;
  }
  __shared__ float sr[CC], si[CC];
  sr[o] = Xr[rowbase + o];
  si[o] = Xi[rowbase + o];
  __syncthreads();

  float Sr = 0.f, Si = 0.f;
  const float* wr = Wr + (size_t)i * CC * CC + o;
  const float* wi = Wi + (size_t)i * CC * CC + o;
  for (int c = 0; c < CC; ++c) {
    const float xr = sr[c], xi = si[c];
    const float wrv = wr[(size_t)c * CC], wiv = wi[(size_t)c * CC];
    Sr += xr * wrv - xi * wiv;
    Si += xr * wiv + xi * wrv;
  }
  const float tr = tmod[((size_t)b * MP + i) * 2 + 0];
  const float ti = tmod[((size_t)b * MP + i) * 2 + 1];
  const float s = (i == 0 ? 1.f : 2.f) * (1.f / (float)NN);
  Xr[rowbase + o] = s * (tr * Sr - ti * Si);
  Xi[rowbase + o] = s * (tr * Si + ti * Sr);
}

// ---------------------------------------------------------------------------
// Kernel 4: inverse truncated DFT (Hermitian doubling folded into Z):
//   out[b,n,o] = sum_i Zr*cos(theta) + Zi*(-sin(theta)),  theta = 2*pi*n*i/N
// Same tiling with n as the M dimension; Zr/Zi chunks staged transposed+padded
// in LDS so B pairs are single ds_load_b64; A from rotation recurrence.
// ---------------------------------------------------------------------------
__global__ __launch_bounds__(128) void inv_dft_kernel(
    const float* __restrict__ Zr, const float* __restrict__ Zi,
    float* __restrict__ out) {
  __shared__ float zrt[CC * LSTR], zit[CC * LSTR];  // 34.8 KB
  const int b = blockIdx.y;
  const int wave = threadIdx.x >> 5;
  const int lane = threadIdx.x & 31;
  const int row = lane & 15;
  const int half = lane >> 4;
  const int n_tile = blockIdx.x * 64 + wave * 16;
  const unsigned nrow = (unsigned)(n_tile + row);

  float wr, wi;
  eneg(nrow * 4u, &wr, &wi);   // rotation per K4 step

  v8f acc[4];
#pragma unroll
  for (int ct = 0; ct < 4; ++ct)
    acc[ct] = (v8f){0.f, 0.f, 0.f, 0.f, 0.f, 0.f, 0.f, 0.f};

  for (int i0 = 0; i0 < MP; i0 += 64) {
    __syncthreads();
    const float4* srcr = (const float4*)Zr + ((size_t)b * MP + i0) * CC / 4;
    const float4* srci = (const float4*)Zi + ((size_t)b * MP + i0) * CC / 4;
    for (int t = threadIdx.x; t < 64 * CC / 4; t += 128) {
      float4 vr = srcr[t];
      float4 vi = srci[t];
      const int n = t >> 4;          // i index within chunk
      const int c = (t & 15) * 4;    // o index
      zrt[(c + 0) * LSTR + n] = vr.x;
      zrt[(c + 1) * LSTR + n] = vr.y;
      zrt[(c + 2) * LSTR + n] = vr.z;
      zrt[(c + 3) * LSTR + n] = vr.w;
      zit[(c + 0) * LSTR + n] = vi.x;
      zit[(c + 1) * LSTR + n] = vi.y;
      zit[(c + 2) * LSTR + n] = vi.z;
      zit[(c + 3) * LSTR + n] = vi.w;
    }
    __syncthreads();

    float u0r, u0i, u1r, u1i;
    eneg(nrow * (unsigned)(i0 + 2 * half), &u0r, &u0i);
    eneg(nrow * (unsigned)(i0 + 2 * half + 1), &u1r, &u1i);

    for (int kk = 0; kk < 64; kk += 4) {
      const v2f ac = {u0r, u1r};    // cos
      const v2f as = {u0i, u1i};    // -sin
      const int il = kk + 2 * half;
#pragma unroll
      for (int ct = 0; ct < 4; ++ct) {
        const float2 pr = *(const float2*)(zrt + (ct * 16 + row) * LSTR + il);
        const float2 pi = *(const float2*)(zit + (ct * 16 + row) * LSTR + il);
        const v2f br = {pr.x, pr.y};
        const v2f bi = {pi.x, pi.y};
        acc[ct] = wmma_f32(ac, br, acc[ct]);
        acc[ct] = wmma_f32(as, bi, acc[ct]);
      }
      const float t0r = u0r * wr - u0i * wi, t0i = u0r * wi + u0i * wr;
      const float t1r = u1r * wr - u1i * wi, t1i = u1r * wi + u1i * wr;
      u0r = t0r; u0i = t0i; u1r = t1r; u1i = t1i;
    }
  }

  const int ng = n_tile + half * 8;
#pragma unroll
  for (int ct = 0; ct < 4; ++ct)
#pragma unroll
    for (int r = 0; r < 8; ++r)
      out[((size_t)b * NN + ng + r) * CC + ct * 16 + row] = acc[ct][r];
}

// ---------------------------------------------------------------------------
extern "C" void kernel_launch(void* const* d_in, const int* in_sizes, int n_in,
                              void* d_out, int out_size, void* d_ws,
                              size_t ws_size, hipStream_t stream) {
  (void)in_sizes; (void)n_in; (void)out_size; (void)ws_size;
  const float* x     = (const float*)d_in[0];  // (16, 16384, 64)
  const float* t_emb = (const float*)d_in[1];  // (16, 256)
  const float* Wr    = (const float*)d_in[2];  // (2049, 64, 64)
  const float* Wi    = (const float*)d_in[3];  // (2049, 64, 64)
  const float* dr    = (const float*)d_in[4];  // (2049, 256)
  const float* di    = (const float*)d_in[5];  // (2049, 256)
  float* out = (float*)d_out;                  // (16, 16384, 64)

  float* ws = (float*)d_ws;
  const size_t xf_elems = (size_t)BB * MP * CC;  // 2,162,688 floats
  float* XFr  = ws;                  // forward spectrum (real), later Zr in place
  float* XFi  = ws + xf_elems;       // forward spectrum (imag), later Zi in place
  float* tmod = ws + 2 * xf_elems;   // (16, 2112) complex interleaved
  // total workspace: (2*xf_elems + BB*MP*2) * 4 bytes ~= 17.6 MB

  tmod_kernel<<<dim3(MP), dim3(64), 0, stream>>>(t_emb, dr, di, tmod);
  fwd_dft_kernel<<<dim3(MP / 64, BB), dim3(128), 0, stream>>>(x, XFr, XFi);
  modemix_kernel<<<dim3(MP, BB), dim3(64), 0, stream>>>(Wr, Wi, tmod, XFr, XFi);
  inv_dft_kernel<<<dim3(NN / 64, BB), dim3(128), 0, stream>>>(XFr, XFi, out);
}